// CompositionNetwork_2877628088537
// MI455X (gfx1250) — compile-verified
//
#include <hip/hip_runtime.h>
#include <hip/hip_bf16.h>

// ---------------------------------------------------------------------------
// Roost-style composition network on MI455X (gfx1250, wave32).
// All dense GEMMs run through V_WMMA_F32_16X16X4_F32.
// Fusions: pair-gather into GEMM1 A-load, BN stats into GEMM1 epilogue
// (f32 global atomics), BN-normalize+SiLU into GEMM2 A-load.
// KDIM/NDIM are compile-time so all B loads use immediate offsets off one
// base pointer and the K loop fully unrolls (cuts VALU address arithmetic).
// ---------------------------------------------------------------------------

typedef float v2f __attribute__((ext_vector_type(2)));
typedef float v8f __attribute__((ext_vector_type(8)));

#define N_NODES   20480
#define FEA       64
#define EMB_LEN   200
#define K_ELEMS   5
#define M_EDGES   (N_NODES * K_ELEMS)   // 102400
#define HID       256
#define N_CRY     4096
#define BN_EPS    1e-5f
#define KC        32                    // K-chunk staged in LDS

enum { AMODE_PAIR = 0, AMODE_PLAIN = 1, AMODE_BNSILU = 2 };

// ---------------------------------------------------------------------------
// WMMA fp32 GEMM: C[M,NDIM] = f(A)[M,KDIM] @ B[KDIM,NDIM] + bias (+ stats).
//  - blockDim.x = 32 * (NDIM/16); one wave per 16-col tile, 16 rows per block
//  - grid.x = Mrows/16 (exact, so EXEC is all-ones at every WMMA)
//  - A lane layout: m = lane&15, k-pair = (lane>>4)*2   (ISA 16x4 f32)
//  - B lane layout: n = lane&15, k-pair = (lane>>4)*2
//  - C lane layout: VGPR i -> row = i + 8*(lane>>4), col = lane&15
// ---------------------------------------------------------------------------
template <int AMODE, bool STATS, int KDIM, int NDIM>
__global__ __launch_bounds__((NDIM / 16) * 32) void gemm_wmma_kernel(
    const float* __restrict__ A,        // PLAIN / BNSILU source (Mrows x KDIM)
    const float* __restrict__ B,        // (KDIM x NDIM) row-major
    const float* __restrict__ bias,     // (NDIM)
    const float* __restrict__ fea,      // PAIR mode node features (N x 64)
    const int*   __restrict__ selfIdx,  // PAIR mode
    const int*   __restrict__ nbrIdx,   // PAIR mode
    const float* __restrict__ bnScale,  // BNSILU mode (KDIM)
    const float* __restrict__ bnShift,  // BNSILU mode (KDIM)
    float* __restrict__ C,              // (Mrows x NDIM)
    float* __restrict__ stats)          // [2*NDIM]: col sums, col sumsq
{
    constexpr int BLOCK = (NDIM / 16) * 32;
    constexpr int NKC   = KDIM / KC;
    constexpr int STEPS = (16 * KC) / BLOCK;   // A-stage elements per thread

    __shared__ float As[16][KC + 1];
    const int lane  = threadIdx.x & 31;
    const int wave  = threadIdx.x >> 5;
    const int row0  = blockIdx.x * 16;
    const int col0  = wave * 16;
    const int n     = lane & 15;           // also the A row index m
    const int khalf = (lane >> 4) * 2;     // 0 or 2

    // One base pointer for all B loads; every access below is base + const.
    const float* __restrict__ Bcol = B + (size_t)khalf * NDIM + col0 + n;

    v8f c = {};

    #pragma unroll
    for (int kc0 = 0; kc0 < NKC; ++kc0) {
        const int kc = kc0 * KC;
        __syncthreads();
        // Cooperative stage of transformed A chunk (16 x KC) into LDS.
        #pragma unroll
        for (int t = 0; t < STEPS; ++t) {
            const int idx = threadIdx.x + t * BLOCK;
            const int r   = idx / KC;
            const int kk  = idx % KC;
            const int gr  = row0 + r;
            const int gk  = kc + kk;
            float v;
            if (AMODE == AMODE_PAIR) {
                const int s  = selfIdx[gr];
                const int nb = nbrIdx[gr];
                v = (gk < FEA) ? fea[s * FEA + gk] : fea[nb * FEA + (gk - FEA)];
            } else if (AMODE == AMODE_PLAIN) {
                v = A[(size_t)gr * KDIM + gk];
            } else { // BNSILU: silu(h*scale + shift)
                const float h = A[(size_t)gr * KDIM + gk];
                const float x = h * bnScale[gk] + bnShift[gk];
                v = x / (1.f + __expf(-x));
            }
            As[r][kk] = v;
        }
        __syncthreads();

        #pragma unroll
        for (int kk = 0; kk < KC; kk += 4) {
            v2f a, b;
            a.x = As[n][kk + khalf];
            a.y = As[n][kk + khalf + 1];
            b.x = Bcol[(size_t)(kc + kk) * NDIM];        // imm-offset loads
            b.y = Bcol[(size_t)(kc + kk + 1) * NDIM];
            c = __builtin_amdgcn_wmma_f32_16x16x4_f32(
                    false, a, false, b, (short)0, c, false, false);
        }
    }

    const float bv = bias[col0 + n];
    #pragma unroll
    for (int i = 0; i < 8; ++i) c[i] += bv;

    const int rbase = row0 + 8 * (lane >> 4);
    float* __restrict__ Cp = C + (size_t)rbase * NDIM + col0 + n;
    #pragma unroll
    for (int i = 0; i < 8; ++i)
        Cp[(size_t)i * NDIM] = c[i];

    if (STATS) {  // per-column partial sums for BatchNorm batch statistics
        float s1 = 0.f, s2 = 0.f;
        #pragma unroll
        for (int i = 0; i < 8; ++i) { s1 += c[i]; s2 += c[i] * c[i]; }
        s1 += __shfl_xor(s1, 16, 32);   // combine the two 8-row halves (same col)
        s2 += __shfl_xor(s2, 16, 32);
        if (lane < 16) {
            atomicAdd(&stats[col0 + n],        s1);
            atomicAdd(&stats[NDIM + col0 + n], s2);
        }
    }
}

// stats -> fused BN scale/shift:  x_norm = h*scale + shift
__global__ void bn_finalize_kernel(const float* __restrict__ stats,
                                   const float* __restrict__ g,
                                   const float* __restrict__ beta,
                                   float* __restrict__ scale,
                                   float* __restrict__ shift,
                                   int Ndim, float invM)
{
    const int i = blockIdx.x * blockDim.x + threadIdx.x;
    if (i >= Ndim) return;
    const float mu  = stats[i] * invM;
    const float var = stats[Ndim + i] * invM - mu * mu;   // biased variance
    const float rs  = rsqrtf(var + BN_EPS);
    const float sc  = rs * g[i];
    scale[i] = sc;
    shift[i] = beta[i] - mu * sc;
}

// gate head output: silu(bn(H)) @ w(256) + b  -> one scalar per row (wave/row)
__global__ void gate_out_kernel(const float* __restrict__ H,
                                const float* __restrict__ scale,
                                const float* __restrict__ shift,
                                const float* __restrict__ w,
                                const float* __restrict__ b1,
                                float* __restrict__ out, int Mrows)
{
    const int lane = threadIdx.x & 31;
    const int wave = threadIdx.x >> 5;
    const int e = blockIdx.x * (blockDim.x >> 5) + wave;
    if (e >= Mrows) return;
    float acc = 0.f;
    #pragma unroll
    for (int i = 0; i < 8; ++i) {
        const int k = lane + 32 * i;
        const float x = H[(size_t)e * HID + k] * scale[k] + shift[k];
        acc += (x / (1.f + __expf(-x))) * w[k];
    }
    #pragma unroll
    for (int off = 16; off; off >>= 1) acc += __shfl_xor(acc, off, 32);
    if (lane == 0) out[e] = acc + b1[0];
}

// graph attention pool over contiguous 5-edge segments (one wave per node)
__global__ void attn_graph_kernel(const float* __restrict__ gate,
                                  const float* __restrict__ msg,
                                  const float* __restrict__ elem_w,
                                  const int*   __restrict__ nbrIdx,
                                  const float* __restrict__ powp,
                                  float* __restrict__ acc,
                                  int nNodes, int first)
{
    const int lane = threadIdx.x & 31;
    const int wave = threadIdx.x >> 5;
    const int node = blockIdx.x * (blockDim.x >> 5) + wave;
    if (node >= nNodes) return;
    const float pv = powp[0];
    const int e0 = node * K_ELEMS;
    float gj[K_ELEMS], aj[K_ELEMS];
    float gmax = -1e30f;
    #pragma unroll
    for (int j = 0; j < K_ELEMS; ++j) { gj[j] = gate[e0 + j]; gmax = fmaxf(gmax, gj[j]); }
    float den = 1e-10f;
    #pragma unroll
    for (int j = 0; j < K_ELEMS; ++j) {
        const float w = elem_w[nbrIdx[e0 + j]];
        aj[j] = powf(w, pv) * __expf(gj[j] - gmax);
        den += aj[j];
    }
    const float inv = 1.f / den;
    #pragma unroll
    for (int h = 0; h < 2; ++h) {
        const int f = lane + 32 * h;
        float v = 0.f;
        #pragma unroll
        for (int j = 0; j < K_ELEMS; ++j)
            v += aj[j] * msg[(size_t)(e0 + j) * FEA + f];
        v *= inv;
        if (first) acc[(size_t)node * FEA + f] = v;
        else       acc[(size_t)node * FEA + f] += v;
    }
}

// crystal attention pool over contiguous 5-node segments; writes mean-of-heads
__global__ void attn_cry_kernel(const float* __restrict__ gate,
                                const float* __restrict__ msg,
                                const float* __restrict__ elem_w,
                                const float* __restrict__ powp,
                                float* __restrict__ out,
                                int nCry, int first)
{
    const int lane = threadIdx.x & 31;
    const int wave = threadIdx.x >> 5;
    const int cry = blockIdx.x * (blockDim.x >> 5) + wave;
    if (cry >= nCry) return;
    const float pv = powp[0];
    const int n0 = cry * K_ELEMS;
    float gj[K_ELEMS], aj[K_ELEMS];
    float gmax = -1e30f;
    #pragma unroll
    for (int j = 0; j < K_ELEMS; ++j) { gj[j] = gate[n0 + j]; gmax = fmaxf(gmax, gj[j]); }
    float den = 1e-10f;
    #pragma unroll
    for (int j = 0; j < K_ELEMS; ++j) {
        const float w = elem_w[n0 + j];
        aj[j] = powf(w, pv) * __expf(gj[j] - gmax);
        den += aj[j];
    }
    const float inv = (1.f / den) * (1.f / 3.f);   // fold mean over 3 heads
    #pragma unroll
    for (int h = 0; h < 2; ++h) {
        const int f = lane + 32 * h;
        float v = 0.f;
        #pragma unroll
        for (int j = 0; j < K_ELEMS; ++j)
            v += aj[j] * msg[(size_t)(n0 + j) * FEA + f];
        v *= inv;
        if (first) out[(size_t)cry * FEA + f] = v;
        else       out[(size_t)cry * FEA + f] += v;
    }
}

// fea += mean(head accum)
__global__ void fea_update_kernel(float* __restrict__ fea,
                                  const float* __restrict__ acc, int total)
{
    const int i = blockIdx.x * blockDim.x + threadIdx.x;
    if (i < total) fea[i] += acc[i] * (1.f / 3.f);
}

// initial embedding: fea[:, :63] = elem_fea @ W + b ; fea[:, 63] = weights
__global__ void embed_kernel(const float* __restrict__ elem_fea,
                             const float* __restrict__ W,
                             const float* __restrict__ b,
                             const float* __restrict__ ew,
                             float* __restrict__ fea)
{
    __shared__ float row[EMB_LEN];
    const int n = blockIdx.x;
    for (int k = threadIdx.x; k < EMB_LEN; k += 64)
        row[k] = elem_fea[(size_t)n * EMB_LEN + k];
    __syncthreads();
    const int c = threadIdx.x;
    if (c < FEA - 1) {
        float acc = b[c];
        for (int k = 0; k < EMB_LEN; ++k) acc += row[k] * W[k * (FEA - 1) + c];
        fea[(size_t)n * FEA + c] = acc;
    } else if (c == FEA - 1) {
        fea[(size_t)n * FEA + c] = ew[n];
    }
}

__global__ void zero_kernel(float* __restrict__ p, int n)
{
    const int i = blockIdx.x * blockDim.x + threadIdx.x;
    if (i < n) p[i] = 0.f;
}

// ---------------------------------------------------------------------------
// Host-side orchestration.
// Input layout (top-level setup_inputs() insertion order; params flattened
// JAX-style with sorted dict keys at each level):
//   0: elem_weights (N,1)   1: elem_fea (N,200)
//   2..40:  cry pools p=0..2, 13 leaves each:
//           [gW, gb, gBeta, gG, gOutW, gOutb, mW, mb, mBeta, mG, mOutW, mOutb, pow]
//   41: emb.W (200x63)  42: emb.b (63)
//   43..159: graphs g=0..2, heads h=0..2, 13 leaves each (same order, din=128)
//   160: self_fea_idx (M)  161: nbr_fea_idx (M)  162: cry_elem_idx (N)  163: n_crystals
// ---------------------------------------------------------------------------
extern "C" void kernel_launch(void* const* d_in, const int* in_sizes, int n_in,
                              void* d_out, int out_size, void* d_ws, size_t ws_size,
                              hipStream_t stream)
{
    (void)in_sizes; (void)n_in; (void)out_size; (void)ws_size;

    const float* elem_weights = (const float*)d_in[0];
    const float* elem_fea     = (const float*)d_in[1];
    const int CRY0 = 2, EMB_W = 41, EMB_B = 42, G0 = 43;
    const int*   selfIdx = (const int*)d_in[160];
    const int*   nbrIdx  = (const int*)d_in[161];
    auto P = [&](int i) { return (const float*)d_in[i]; };

    // workspace carve-up (256B aligned); requires ~142 MB
    char* w = (char*)d_ws;
    size_t off = 0;
    auto alloc = [&](size_t bytes) {
        float* p = (float*)(w + off);
        off += (bytes + 255) & ~(size_t)255;
        return p;
    };
    float* fea     = alloc((size_t)N_NODES * FEA * 4);   //  5.2 MB
    float* headAcc = alloc((size_t)N_NODES * FEA * 4);   //  5.2 MB
    float* H       = alloc((size_t)M_EDGES * HID * 4);   // 104.9 MB (shared gate/msg)
    float* msgOut  = alloc((size_t)M_EDGES * FEA * 4);   // 26.2 MB
    float* gateOut = alloc((size_t)M_EDGES * 4);         //  0.4 MB
    float* stats   = alloc(2 * HID * 4);
    float* bnScale = alloc(HID * 4);
    float* bnShift = alloc(HID * 4);

    // ---- embedding ----
    embed_kernel<<<N_NODES, 64, 0, stream>>>(elem_fea, P(EMB_W), P(EMB_B),
                                             elem_weights, fea);

    // ---- 3 message-passing layers x 3 heads ----
    for (int g = 0; g < 3; ++g) {
        for (int h = 0; h < 3; ++h) {
            const int base = G0 + (g * 3 + h) * 13;
            const float *gW = P(base+0), *gb = P(base+1), *gBeta = P(base+2), *gG = P(base+3);
            const float *gOW = P(base+4), *gOb = P(base+5);
            const float *mW = P(base+6), *mb = P(base+7), *mBeta = P(base+8), *mG = P(base+9);
            const float *mOW = P(base+10), *mOb = P(base+11), *pw = P(base+12);

            // gate MLP: GEMM1 (gathered pair @ W) + BN stats -> bnScale/shift -> dot
            zero_kernel<<<2, 256, 0, stream>>>(stats, 2 * HID);
            gemm_wmma_kernel<AMODE_PAIR, true, 2 * FEA, HID>
                <<<M_EDGES / 16, 512, 0, stream>>>(
                nullptr, gW, gb, fea, selfIdx, nbrIdx, nullptr, nullptr, H, stats);
            bn_finalize_kernel<<<1, 256, 0, stream>>>(stats, gG, gBeta, bnScale,
                                                      bnShift, HID, 1.f / M_EDGES);
            gate_out_kernel<<<M_EDGES / 8, 256, 0, stream>>>(H, bnScale, bnShift,
                                                             gOW, gOb, gateOut, M_EDGES);

            // msg MLP: GEMM1 + stats, then BN/SiLU fused into GEMM2 A-load
            zero_kernel<<<2, 256, 0, stream>>>(stats, 2 * HID);
            gemm_wmma_kernel<AMODE_PAIR, true, 2 * FEA, HID>
                <<<M_EDGES / 16, 512, 0, stream>>>(
                nullptr, mW, mb, fea, selfIdx, nbrIdx, nullptr, nullptr, H, stats);
            bn_finalize_kernel<<<1, 256, 0, stream>>>(stats, mG, mBeta, bnScale,
                                                      bnShift, HID, 1.f / M_EDGES);
            gemm_wmma_kernel<AMODE_BNSILU, false, HID, FEA>
                <<<M_EDGES / 16, 128, 0, stream>>>(
                H, mOW, mOb, nullptr, nullptr, nullptr, bnScale, bnShift,
                msgOut, nullptr);

            attn_graph_kernel<<<N_NODES / 8, 256, 0, stream>>>(
                gateOut, msgOut, elem_weights, nbrIdx, pw, headAcc, N_NODES, h == 0);
        }
        fea_update_kernel<<<(N_NODES * FEA + 255) / 256, 256, 0, stream>>>(
            fea, headAcc, N_NODES * FEA);
    }

    // ---- crystal pooling: 3 heads, din = 64 ----
    for (int p = 0; p < 3; ++p) {
        const int base = CRY0 + p * 13;
        const float *gW = P(base+0), *gb = P(base+1), *gBeta = P(base+2), *gG = P(base+3);
        const float *gOW = P(base+4), *gOb = P(base+5);
        const float *mW = P(base+6), *mb = P(base+7), *mBeta = P(base+8), *mG = P(base+9);
        const float *mOW = P(base+10), *mOb = P(base+11), *pw = P(base+12);

        zero_kernel<<<2, 256, 0, stream>>>(stats, 2 * HID);
        gemm_wmma_kernel<AMODE_PLAIN, true, FEA, HID>
            <<<N_NODES / 16, 512, 0, stream>>>(
            fea, gW, gb, nullptr, nullptr, nullptr, nullptr, nullptr, H, stats);
        bn_finalize_kernel<<<1, 256, 0, stream>>>(stats, gG, gBeta, bnScale,
                                                  bnShift, HID, 1.f / N_NODES);
        gate_out_kernel<<<N_NODES / 8, 256, 0, stream>>>(H, bnScale, bnShift,
                                                         gOW, gOb, gateOut, N_NODES);

        zero_kernel<<<2, 256, 0, stream>>>(stats, 2 * HID);
        gemm_wmma_kernel<AMODE_PLAIN, true, FEA, HID>
            <<<N_NODES / 16, 512, 0, stream>>>(
            fea, mW, mb, nullptr, nullptr, nullptr, nullptr, nullptr, H, stats);
        bn_finalize_kernel<<<1, 256, 0, stream>>>(stats, mG, mBeta, bnScale,
                                                  bnShift, HID, 1.f / N_NODES);
        gemm_wmma_kernel<AMODE_BNSILU, false, HID, FEA>
            <<<N_NODES / 16, 128, 0, stream>>>(
            H, mOW, mOb, nullptr, nullptr, nullptr, bnScale, bnShift,
            msgOut, nullptr);

        attn_cry_kernel<<<N_CRY / 8, 256, 0, stream>>>(
            gateOut, msgOut, elem_weights, pw, (float*)d_out, N_CRY, p == 0);
    }
}